// MAGPoolGCNLayer_39865886442007
// MI455X (gfx1250) — compile-verified
//
#include <hip/hip_runtime.h>
#include <hip/hip_bf16.h>

typedef float v2f __attribute__((ext_vector_type(2)));
typedef float v8f __attribute__((ext_vector_type(8)));

#define D_IN   256
#define SUB_IN 64
#define N_OUT  256   // HEADS * SUB_H

// ---------------- Kernel 1: deg[i] = 1.0 (self loop) ----------------
__global__ __launch_bounds__(256) void k_init_deg(float* __restrict__ deg, int n) {
  int i = blockIdx.x * 256 + threadIdx.x;
  if (i < n) deg[i] = 1.0f;
}

// ---------------- Kernel 2: deg[col[e]] += 1 ----------------
__global__ __launch_bounds__(256) void k_count_deg(const int* __restrict__ ei,
                                                   float* __restrict__ deg,
                                                   int n_edges) {
  int e = blockIdx.x * 256 + threadIdx.x;
  if (e >= n_edges) return;
  int col = ei[n_edges + e];
  unsafeAtomicAdd(&deg[col], 1.0f);   // hardware global_atomic_add_f32
}

// ------- Kernel 3: dinv[i] = rsqrt(deg); agg[i][f] = dinv^2 * x0[i][f] -------
__global__ __launch_bounds__(256) void k_init_agg(const float* __restrict__ x,
                                                  const float* __restrict__ deg,
                                                  float* __restrict__ dinv,
                                                  float* __restrict__ agg,
                                                  int n) {
  int gid = blockIdx.x * 256 + threadIdx.x;
  int node = gid >> 6;           // 64 threads per node
  int f    = gid & 63;
  if (node >= n) return;
  float di = __frsqrt_rn(deg[node]);   // deg >= 1 always
  if (f == 0) dinv[node] = di;
  agg[node * SUB_IN + f] = di * di * x[(size_t)node * D_IN + f];
}

// ------- Kernel 4: one wave per edge; agg[col] += norm * x0[row] -------
__global__ __launch_bounds__(256) void k_scatter(const int* __restrict__ ei,
                                                 const float* __restrict__ x,
                                                 const float* __restrict__ dinv,
                                                 float* __restrict__ agg,
                                                 int n_edges) {
  int gid  = blockIdx.x * 256 + threadIdx.x;
  int e    = gid >> 5;           // wave32: one wave per edge
  if (e >= n_edges) return;
  int lane = gid & 31;
  int row  = ei[e];
  int col  = ei[n_edges + e];
  float norm = dinv[row] * dinv[col];
  const float* xr = x + (size_t)row * D_IN;   // coalesced 128B per half
  float* ac = agg + (size_t)col * SUB_IN;
  unsafeAtomicAdd(ac + lane,      xr[lane]      * norm);
  unsafeAtomicAdd(ac + 32 + lane, xr[32 + lane] * norm);
}

// ------- Kernel 5: out = relu(agg @ Wcat + b), written twice (x_cat, heads) -------
// agg: n x 64, Wcat[k][col] = W[col>>6][k][col&63], out tile = 16x16 per wave.
__global__ __launch_bounds__(256) void k_gemm_wmma(const float* __restrict__ agg,
                                                   const float* __restrict__ W,
                                                   const float* __restrict__ bias,
                                                   float* __restrict__ out,
                                                   int n, int total_tiles) {
  const int tid  = threadIdx.x;
  const int lane = tid & 31;
  const int tile = blockIdx.x * 8 + (tid >> 5);   // 8 waves/block
  if (tile >= total_tiles) return;
  const int mt = tile >> 4;          // N_OUT/16 == 16 column tiles
  const int nt = tile & 15;
  const int m0 = mt * 16;
  const int n0 = nt * 16;
  const int half = lane >> 4;        // 0: K%4 in {0,1}; 1: K%4 in {2,3}
  const int l15  = lane & 15;

  const int colN = n0 + l15;                               // 0..255
  const float* Wcol = W + (colN >> 6) * (SUB_IN * 64) + (colN & 63);

  int rA = m0 + l15; if (rA > n - 1) rA = n - 1;           // clamp pad rows
  const float* arow = agg + (size_t)rA * SUB_IN + (half << 1);

  v8f c = {};
#pragma unroll
  for (int kk = 0; kk < 16; ++kk) {                        // K = 64 = 16 * 4
    const int k0 = kk * 4;
    v2f a = *(const v2f*)(arow + k0);                      // A[M][k0+2h .. +1]
    const int kb = k0 + (half << 1);
    v2f b;
    b.x = Wcol[(size_t)kb * 64];                           // B[kb  ][colN]
    b.y = Wcol[(size_t)(kb + 1) * 64];                     // B[kb+1][colN]
    c = __builtin_amdgcn_wmma_f32_16x16x4_f32(
            /*neg_a=*/false, a, /*neg_b=*/false, b,
            /*c_mod=*/(short)0, c, /*reuse_a=*/false, /*reuse_b=*/false);
  }

  const float bv = bias[colN];
  const size_t off2 = (size_t)n * N_OUT;                   // heads copy
#pragma unroll
  for (int r = 0; r < 8; ++r) {
    int row = m0 + r + (half << 3);                        // C layout: M=r / r+8
    if (row < n) {
      float v = c[r] + bv;
      v = v > 0.0f ? v : 0.0f;
      size_t o = (size_t)row * N_OUT + colN;
      out[o]        = v;   // x_cat
      out[off2 + o] = v;   // heads (identical data)
    }
  }
}

extern "C" void kernel_launch(void* const* d_in, const int* in_sizes, int n_in,
                              void* d_out, int out_size, void* d_ws, size_t ws_size,
                              hipStream_t stream) {
  const float* x    = (const float*)d_in[0];
  const int*   ei   = (const int*)d_in[1];
  const float* W    = (const float*)d_in[2];
  const float* bias = (const float*)d_in[3];
  float* out = (float*)d_out;

  const int n       = in_sizes[0] / D_IN;   // 50000
  const int n_edges = in_sizes[1] / 2;      // 1600000

  // workspace layout (floats): deg[n] | dinv[n] | agg[n*64]
  float* deg  = (float*)d_ws;
  float* dinv = deg + n;
  float* agg  = dinv + n;

  k_init_deg<<<(n + 255) / 256, 256, 0, stream>>>(deg, n);
  k_count_deg<<<(n_edges + 255) / 256, 256, 0, stream>>>(ei, deg, n_edges);
  k_init_agg<<<(n * SUB_IN + 255) / 256, 256, 0, stream>>>(x, deg, dinv, agg, n);
  {
    long long threads = (long long)n_edges * 32;
    k_scatter<<<(int)((threads + 255) / 256), 256, 0, stream>>>(ei, x, dinv, agg, n_edges);
  }
  {
    const int mtiles = (n + 15) / 16;           // 3125
    const int total_tiles = mtiles * (N_OUT / 16);
    k_gemm_wmma<<<(total_tiles + 7) / 8, 256, 0, stream>>>(agg, W, bias, out, n, total_tiles);
  }
}